// MyNet_14405320311165
// MI455X (gfx1250) — compile-verified
//
#include <hip/hip_runtime.h>
#include <hip/hip_bf16.h>

typedef float v2f_t __attribute__((ext_vector_type(2)));
typedef float v8f_t __attribute__((ext_vector_type(8)));

#define N_PROT 17185
#define BATCH  32
#define JT     16
#define KC     448                          // k-chunk staged in LDS
#define LDS_STRIDE 452                      // KC+4: rows 16B aligned, bank stride 4 -> conflict free b64
#define WPB    4                            // waves per block (wave32)
#define KSEG   4                            // k-split factor for occupancy
#define NT     ((N_PROT + JT - 1) / JT)     // 1075 j-tiles
#define JPAD   (NT * JT)                    // 17200 padded columns
#define BN_EPS 1e-5f

// ---------------------------------------------------------------- zero ws
__global__ void zero_f32(float* __restrict__ p, int n) {
    int i = blockIdx.x * blockDim.x + threadIdx.x;
    if (i < n) p[i] = 0.0f;
}

// ---------------------------------------------------------- masked GEMM
// h[b, j] = sum_k x[b,k] * (w1[j,k] * adj[j,k])
// One wave computes a 32(batch) x 16(j) tile over one k-segment using
// V_WMMA_F32_16X16X4_F32; partial tiles are atomically accumulated.
__global__ __launch_bounds__(WPB * 32)
void gemm_masked(const float* __restrict__ x,
                 const float* __restrict__ adj,
                 const float* __restrict__ w1,
                 float* __restrict__ hacc)
{
    __shared__ float xs[BATCH * LDS_STRIDE];   // 57856 B

    const int tid  = threadIdx.x;
    const int wave = tid >> 5;
    const int lane = tid & 31;
    const int lh   = lane & 15;
    const int khi  = lane >> 4;               // 0 -> k{0,1}, 1 -> k{2,3}

    const int jtile = blockIdx.x * WPB + wave;
    const int seg   = blockIdx.y;

    const long jbase = (long)jtile * JT;
    const long j     = jbase + lh;
    const long jeff  = (j < N_PROT) ? j : (long)(N_PROT - 1);
    const float* __restrict__ wrow = w1  + jeff * (long)N_PROT;
    const float* __restrict__ arow = adj + jeff * (long)N_PROT;

    v8f_t acc0 = {0.f, 0.f, 0.f, 0.f, 0.f, 0.f, 0.f, 0.f};  // batch 0..15 rows
    v8f_t acc1 = {0.f, 0.f, 0.f, 0.f, 0.f, 0.f, 0.f, 0.f};  // batch 16..31 rows

    const int nchunks = (N_PROT + KC - 1) / KC;   // 39

    for (int c = seg; c < nchunks; c += KSEG) {
        const int k0 = c * KC;

        __syncthreads();
        // cooperative stage of x[0..31][k0 .. k0+KC) into LDS, zero padded past N
        for (int i = tid; i < BATCH * (KC / 4); i += WPB * 32) {
            const int row = i / (KC / 4);
            const int c4  = (i % (KC / 4)) * 4;
            const int kk  = k0 + c4;
            float4 v;
            if (kk + 3 < N_PROT) {
                v = *(const float4*)(x + (long)row * N_PROT + kk);
            } else {
                v.x = (kk + 0 < N_PROT) ? x[(long)row * N_PROT + kk + 0] : 0.f;
                v.y = (kk + 1 < N_PROT) ? x[(long)row * N_PROT + kk + 1] : 0.f;
                v.z = (kk + 2 < N_PROT) ? x[(long)row * N_PROT + kk + 2] : 0.f;
                v.w = (kk + 3 < N_PROT) ? x[(long)row * N_PROT + kk + 3] : 0.f;
            }
            *(float4*)(xs + row * LDS_STRIDE + c4) = v;
        }
        __syncthreads();

        const int kof_base = 2 * khi;
        if (k0 + KC <= N_PROT) {
            // fast path: no k bounds checks
            #pragma unroll 8
            for (int kk = 0; kk < KC; kk += 4) {
                const int kof = kk + kof_base;
                v2f_t a0 = *(const v2f_t*)(xs + lh * LDS_STRIDE + kof);
                v2f_t a1 = *(const v2f_t*)(xs + (lh + 16) * LDS_STRIDE + kof);
                const long kg = (long)k0 + kof;
                v2f_t bf;
                bf.x = wrow[kg]     * arow[kg];
                bf.y = wrow[kg + 1] * arow[kg + 1];
                acc0 = __builtin_amdgcn_wmma_f32_16x16x4_f32(
                    false, a0, false, bf, (short)0, acc0, false, false);
                acc1 = __builtin_amdgcn_wmma_f32_16x16x4_f32(
                    false, a1, false, bf, (short)0, acc1, false, false);
            }
        } else {
            // tail chunk: clamp addresses; xs holds zeros past N so products vanish
            for (int kk = 0; kk < KC; kk += 4) {
                const int kof = kk + kof_base;
                v2f_t a0 = *(const v2f_t*)(xs + lh * LDS_STRIDE + kof);
                v2f_t a1 = *(const v2f_t*)(xs + (lh + 16) * LDS_STRIDE + kof);
                const long kg  = (long)k0 + kof;
                const long kg0 = (kg     < N_PROT) ? kg     : (long)(N_PROT - 1);
                const long kg1 = (kg + 1 < N_PROT) ? kg + 1 : (long)(N_PROT - 1);
                v2f_t bf;
                bf.x = wrow[kg0] * arow[kg0];
                bf.y = wrow[kg1] * arow[kg1];
                acc0 = __builtin_amdgcn_wmma_f32_16x16x4_f32(
                    false, a0, false, bf, (short)0, acc0, false, false);
                acc1 = __builtin_amdgcn_wmma_f32_16x16x4_f32(
                    false, a1, false, bf, (short)0, acc1, false, false);
            }
        }
    }

    // C/D layout: lane(0..15)=column j, VGPR r = row (r + 8*khi); acc1 adds +16
    if (j < N_PROT) {
        #pragma unroll
        for (int r = 0; r < 8; ++r) {
            atomicAdd(&hacc[(long)(r + 8 * khi)      * JPAD + j], acc0[r]);
            atomicAdd(&hacc[(long)(r + 8 * khi + 16) * JPAD + j], acc1[r]);
        }
    }
}

// ------------------------------------------------- fused epilogue (tiny)
// out[b] = BN3( sum_j BN1(relu(h[b,j] + b1[j])) * w3[j] + b3 )
__global__ __launch_bounds__(256)
void finalize(const float* __restrict__ hacc,
              const float* __restrict__ b1,
              const float* __restrict__ g1, const float* __restrict__ be1,
              const float* __restrict__ m1, const float* __restrict__ va1,
              const float* __restrict__ w3, const float* __restrict__ b3,
              const float* __restrict__ g3, const float* __restrict__ be3,
              const float* __restrict__ m3, const float* __restrict__ va3,
              float* __restrict__ out)
{
    __shared__ float red[256];
    const int b = blockIdx.x;
    float s = 0.0f;
    for (int jj = threadIdx.x; jj < N_PROT; jj += 256) {
        float h = hacc[(long)b * JPAD + jj] + b1[jj];
        h = fmaxf(h, 0.0f);
        h = (h - m1[jj]) * rsqrtf(va1[jj] + BN_EPS) * g1[jj] + be1[jj];
        s += h * w3[jj];
    }
    red[threadIdx.x] = s;
    __syncthreads();
    for (int off = 128; off > 0; off >>= 1) {
        if (threadIdx.x < off) red[threadIdx.x] += red[threadIdx.x + off];
        __syncthreads();
    }
    if (threadIdx.x == 0) {
        float h = red[0] + b3[0];
        out[b] = (h - m3[0]) * rsqrtf(va3[0] + BN_EPS) * g3[0] + be3[0];
    }
}

// ---------------------------------------------------------------- launch
extern "C" void kernel_launch(void* const* d_in, const int* in_sizes, int n_in,
                              void* d_out, int out_size, void* d_ws, size_t ws_size,
                              hipStream_t stream) {
    const float* x   = (const float*)d_in[0];
    const float* adj = (const float*)d_in[1];
    const float* w1  = (const float*)d_in[2];
    const float* b1  = (const float*)d_in[3];
    const float* g1  = (const float*)d_in[4];
    const float* be1 = (const float*)d_in[5];
    const float* m1  = (const float*)d_in[6];
    const float* va1 = (const float*)d_in[7];
    const float* w3  = (const float*)d_in[8];
    const float* b3  = (const float*)d_in[9];
    const float* g3  = (const float*)d_in[10];
    const float* be3 = (const float*)d_in[11];
    const float* m3  = (const float*)d_in[12];
    const float* va3 = (const float*)d_in[13];

    float* hacc = (float*)d_ws;          // BATCH * JPAD floats (~2.2 MB)
    float* out  = (float*)d_out;         // 32 floats

    const int nzero = BATCH * JPAD;
    zero_f32<<<(nzero + 255) / 256, 256, 0, stream>>>(hacc, nzero);

    dim3 grid((NT + WPB - 1) / WPB, KSEG);   // (269, 4) -> ~4300 waves
    gemm_masked<<<grid, WPB * 32, 0, stream>>>(x, adj, w1, hacc);

    finalize<<<BATCH, 256, 0, stream>>>(hacc, b1, g1, be1, m1, va1,
                                        w3, b3, g3, be3, m3, va3, out);
}